// InstanceMetricLoss_6665789243927
// MI455X (gfx1250) — compile-verified
//
#include <hip/hip_runtime.h>
#include <hip/hip_bf16.h>

typedef float v2f __attribute__((ext_vector_type(2)));
typedef float v8f __attribute__((ext_vector_type(8)));

#define PHALF 4096        // B/2
#define DDIM  128
#define LSTR  68          // 64 + 4 pad; 68*4 bytes = 272 = 16*17 (float4-aligned rows), 68%64=4 (bank spread)

__global__ __launch_bounds__(256) void iml_zero_kernel(float* R, float* C, float* diag) {
    int i = blockIdx.x * blockDim.x + threadIdx.x;
    if (i < PHALF) { R[i] = 0.f; C[i] = 0.f; diag[i] = 0.f; }
}

// Computes 64x64 tile of G = A_even * A_odd^T, fused exp + masked row/col partial sums.
__global__ __launch_bounds__(256) void iml_gram_kernel(const float* __restrict__ X,
                                                       float* __restrict__ R,
                                                       float* __restrict__ C,
                                                       float* __restrict__ diag) {
    __shared__ float Ae[64 * LSTR];   // even rows, one 64-wide K chunk
    __shared__ float Ao[64 * LSTR];   // odd rows,  one 64-wide K chunk
    __shared__ float rowAcc[64];
    __shared__ float colAcc[64];

    const int tid = threadIdx.x;
    const int bm = blockIdx.x;        // row block of G (even rows of input)
    const int bn = blockIdx.y;        // col block of G (odd rows of input)

    if (tid < 64) rowAcc[tid] = 0.f;
    else if (tid < 128) colAcc[tid - 64] = 0.f;

    const int lane = tid & 31;
    const int wave = tid >> 5;
    const int half = lane >> 4;       // K-half / M-half selector per WMMA layout
    const int rrow = lane & 15;       // fragment row (A) == fragment col (B) == C column

    const int tm  = wave >> 1;        // 16-row tile index within 64-row block
    const int tn0 = (wave & 1) * 2;   // two adjacent 16-col tiles per wave
    const int tn1 = tn0 + 1;

    v8f acc0 = {0.f,0.f,0.f,0.f,0.f,0.f,0.f,0.f};
    v8f acc1 = {0.f,0.f,0.f,0.f,0.f,0.f,0.f,0.f};

    const float* arow  = &Ae[(tm  * 16 + rrow) * LSTR + 2 * half];
    const float* brow0 = &Ao[(tn0 * 16 + rrow) * LSTR + 2 * half];
    const float* brow1 = &Ao[(tn1 * 16 + rrow) * LSTR + 2 * half];

    for (int kc = 0; kc < DDIM; kc += 64) {
        __syncthreads();   // previous chunk fully consumed
        // stage 64 rows x 64 floats of each operand: 1024 float4 per matrix, 4 per thread
        for (int i = 0; i < 4; ++i) {
            int e  = tid + i * 256;            // 0..1023
            int r  = e >> 4;                   // row 0..63
            int c4 = e & 15;                   // float4 column within chunk
            const float4* gsa = (const float4*)(X + (size_t)(2 * (bm * 64 + r)) * DDIM + kc);
            const float4* gsb = (const float4*)(X + (size_t)(2 * (bn * 64 + r) + 1) * DDIM + kc);
            ((float4*)&Ae[r * LSTR])[c4] = gsa[c4];
            ((float4*)&Ao[r * LSTR])[c4] = gsb[c4];
        }
        __syncthreads();

        // 16 K-steps of V_WMMA_F32_16X16X4_F32 per tile; A fragment reused for both tiles
        for (int kk = 0; kk < 64; kk += 4) {
            v2f a  = *(const v2f*)(arow  + kk);
            v2f b0 = *(const v2f*)(brow0 + kk);
            v2f b1 = *(const v2f*)(brow1 + kk);
            acc0 = __builtin_amdgcn_wmma_f32_16x16x4_f32(false, a, false, b0,
                                                         (short)0, acc0, false, false);
            acc1 = __builtin_amdgcn_wmma_f32_16x16x4_f32(false, a, false, b1,
                                                         (short)0, acc1, false, false);
        }
    }

    // Epilogue: exp(1 + G/128), exclusion masks, partial row/col sums.
    // C layout: VGPR r -> M = r + 8*(lane>>4), N = lane&15.
    {
        // tile 0
        int gn = bn * 64 + tn0 * 16 + rrow;
        float colSum = 0.f;
        for (int r2 = 0; r2 < 8; ++r2) {
            int gm = bm * 64 + tm * 16 + r2 + 8 * half;
            float dval = acc0[r2] * (1.0f / 128.0f);
            float ev = __expf(1.0f + dval);
            float rowInc = (gn == 2 * gm + 1) ? 0.f : ev;
            float colInc = (gm == 2 * gn)     ? 0.f : ev;
            colSum += colInc;
            atomicAdd(&rowAcc[tm * 16 + r2 + 8 * half], rowInc);
            if (gm == gn) diag[gm] = dval;
        }
        atomicAdd(&colAcc[tn0 * 16 + rrow], colSum);
    }
    {
        // tile 1
        int gn = bn * 64 + tn1 * 16 + rrow;
        float colSum = 0.f;
        for (int r2 = 0; r2 < 8; ++r2) {
            int gm = bm * 64 + tm * 16 + r2 + 8 * half;
            float dval = acc1[r2] * (1.0f / 128.0f);
            float ev = __expf(1.0f + dval);
            float rowInc = (gn == 2 * gm + 1) ? 0.f : ev;
            float colInc = (gm == 2 * gn)     ? 0.f : ev;
            colSum += colInc;
            atomicAdd(&rowAcc[tm * 16 + r2 + 8 * half], rowInc);
            if (gm == gn) diag[gm] = dval;
        }
        atomicAdd(&colAcc[tn1 * 16 + rrow], colSum);
    }

    __syncthreads();
    if (tid < 64)       atomicAdd(&R[bm * 64 + tid],        rowAcc[tid]);
    else if (tid < 128) atomicAdd(&C[bn * 64 + (tid - 64)], colAcc[tid - 64]);
}

__global__ __launch_bounds__(256) void iml_final_kernel(const float* __restrict__ R,
                                                        const float* __restrict__ C,
                                                        const float* __restrict__ diag,
                                                        float* __restrict__ out) {
    __shared__ float red[256];
    float s = 0.f;
    for (int p = threadIdx.x; p < PHALF; p += 256) {
        float J = __logf(1e-8f + R[p] + C[p]) - diag[p];
        float r = fmaxf(J, 0.f);
        s += r * r;
    }
    red[threadIdx.x] = s;
    __syncthreads();
    for (int off = 128; off > 0; off >>= 1) {
        if (threadIdx.x < off) red[threadIdx.x] += red[threadIdx.x + off];
        __syncthreads();
    }
    if (threadIdx.x == 0) out[0] = red[0] * (1.0f / 8192.0f);
}

extern "C" void kernel_launch(void* const* d_in, const int* in_sizes, int n_in,
                              void* d_out, int out_size, void* d_ws, size_t ws_size,
                              hipStream_t stream) {
    const float* X = (const float*)d_in[0];     // 8192 x 128 fp32
    float* R    = (float*)d_ws;                 // 4096 floats: masked row sums of exp
    float* C    = R + PHALF;                    // 4096 floats: masked col sums of exp
    float* diag = C + PHALF;                    // 4096 floats: G[p,p]/128

    iml_zero_kernel<<<(PHALF + 255) / 256, 256, 0, stream>>>(R, C, diag);
    iml_gram_kernel<<<dim3(64, 64), 256, 0, stream>>>(X, R, C, diag);
    iml_final_kernel<<<1, 256, 0, stream>>>(R, C, diag, (float*)d_out);
}